// DarkChannel_77713138254515
// MI455X (gfx1250) — compile-verified
//
#include <hip/hip_runtime.h>
#include <hip/hip_bf16.h>

#ifndef __has_builtin
#define __has_builtin(x) 0
#endif

// gfx1250 async global->LDS copy path (ASYNCcnt). Guarded so the file always
// compiles; fallback is a plain global_load + ds_store with identical layout.
#if defined(__gfx1250__) && \
    __has_builtin(__builtin_amdgcn_global_load_async_to_lds_b32) && \
    __has_builtin(__builtin_amdgcn_s_wait_asynccnt)
#define DC_ASYNC 1
#else
#define DC_ASYNC 0
#endif

#if DC_ASYNC
typedef __attribute__((address_space(1))) int dc_gint;   // global
typedef __attribute__((address_space(3))) int dc_lint;   // LDS
#endif

#define H      1024
#define W      1024
#define HWSZ   (H * W)
#define CHN    3
#define TILE_X 128
#define TILE_Y 128
#define HALO   7                      // window 15 -> 7 each side
#define RW     (TILE_X + 2 * HALO)    // 142: halo'd region width
#define RH     (TILE_Y + 2 * HALO)    // 142: halo'd region height
#define NTHREADS 256

__device__ __forceinline__ int clampi(int v, int lo, int hi) {
    return v < lo ? lo : (v > hi ? hi : v);
}

__global__ __launch_bounds__(NTHREADS)
void dark_channel_kernel(const float* __restrict__ img, float* __restrict__ out) {
    // 142x142 channel-min region and 142x128 vertical-min buffer.
    __shared__ float s_cmin[RH * RW];     // 80,656 B
    __shared__ float s_vmin[TILE_Y * RW]; // 72,704 B   (total 153,360 B)

    const int tid = threadIdx.x;
    const int x0  = blockIdx.x * TILE_X;
    const int y0  = blockIdx.y * TILE_Y;
    const int b   = blockIdx.z;

    const float* p0 = img + (size_t)b * CHN * HWSZ;
    const float* p1 = p0 + HWSZ;
    const float* p2 = p0 + 2 * HWSZ;

    // ---------------- Phase 1: load halo region, fuse channel-min ----------
    // Replicate-clamp == +inf border for a min filter, and keeps every
    // address in-bounds (safe for async DMA into LDS).
#if DC_ASYNC
    // Issue channel 0 as async global->LDS copies (tracked by ASYNCcnt).
    for (int idx = tid; idx < RH * RW; idx += NTHREADS) {
        const int r  = idx / RW;
        const int c  = idx - r * RW;
        const int gy = clampi(y0 - HALO + r, 0, H - 1);
        const int gx = clampi(x0 - HALO + c, 0, W - 1);
        __builtin_amdgcn_global_load_async_to_lds_b32(
            (dc_gint*)(p0 + (size_t)gy * W + gx),
            (dc_lint*)&s_cmin[idx],
            0, 0);
    }
    // Each lane consumes exactly the LDS word it issued -> wave-local wait
    // on ASYNCcnt is sufficient, no workgroup barrier needed yet.
    __builtin_amdgcn_s_wait_asynccnt(0);
    for (int idx = tid; idx < RH * RW; idx += NTHREADS) {
        const int r  = idx / RW;
        const int c  = idx - r * RW;
        const int gy = clampi(y0 - HALO + r, 0, H - 1);
        const int gx = clampi(x0 - HALO + c, 0, W - 1);
        const size_t g = (size_t)gy * W + gx;
        float v = s_cmin[idx];
        v = fminf(v, p1[g]);
        v = fminf(v, p2[g]);
        s_cmin[idx] = v;
    }
#else
    for (int idx = tid; idx < RH * RW; idx += NTHREADS) {
        const int r  = idx / RW;
        const int c  = idx - r * RW;
        const int gy = clampi(y0 - HALO + r, 0, H - 1);
        const int gx = clampi(x0 - HALO + c, 0, W - 1);
        const size_t g = (size_t)gy * W + gx;
        float v = p0[g];
        v = fminf(v, p1[g]);
        v = fminf(v, p2[g]);
        s_cmin[idx] = v;
    }
#endif
    __syncthreads();

    // ---------------- Phase 2: vertical 15-tap min ------------------------
    // Output rows 0..TILE_Y-1 need cmin rows y..y+14; keep full halo width.
    for (int idx = tid; idx < TILE_Y * RW; idx += NTHREADS) {
        const int y = idx / RW;
        const int x = idx - y * RW;
        float m = s_cmin[y * RW + x];
#pragma unroll
        for (int k = 1; k < 15; ++k) {
            m = fminf(m, s_cmin[(y + k) * RW + x]);
        }
        s_vmin[idx] = m;
    }
    __syncthreads();

    // ---------------- Phase 3: horizontal 15-tap min + NT store -----------
    float* outp = out + (size_t)b * HWSZ;
    for (int idx = tid; idx < TILE_Y * TILE_X; idx += NTHREADS) {
        const int y = idx >> 7;            // / TILE_X
        const int x = idx & (TILE_X - 1);  // % TILE_X
        const float* row = &s_vmin[y * RW + x];
        float m = row[0];
#pragma unroll
        for (int k = 1; k < 15; ++k) {
            m = fminf(m, row[k]);
        }
        // Output is write-once, never re-read: keep it out of L2 so halo
        // re-reads from neighboring tiles stay resident instead.
        __builtin_nontemporal_store(m, &outp[(size_t)(y0 + y) * W + (x0 + x)]);
    }
}

extern "C" void kernel_launch(void* const* d_in, const int* in_sizes, int n_in,
                              void* d_out, int out_size, void* d_ws, size_t ws_size,
                              hipStream_t stream) {
    const float* img = (const float*)d_in[0];
    float* out = (float*)d_out;
    const int batch = in_sizes[0] / (CHN * HWSZ);   // 16 for the reference shapes
    dim3 grid(W / TILE_X, H / TILE_Y, batch);
    dim3 block(NTHREADS);
    dark_channel_kernel<<<grid, block, 0, stream>>>(img, out);
}